// GNN_55001351193097
// MI455X (gfx1250) — compile-verified
//
#include <hip/hip_runtime.h>
#include <math.h>

#define N_NODES 100000
#define N_EDGES 1600000
#define DIN     10
#define C64     64
#define D1      84   // 2*DIN + C64

typedef __attribute__((ext_vector_type(2))) float v2f;
typedef __attribute__((ext_vector_type(8))) float v8f;

// ---- monotone order-preserving float <-> uint for atomic segment-max ----
__device__ __forceinline__ unsigned flip_f(float f) {
  unsigned u = __float_as_uint(f);
  return (u & 0x80000000u) ? ~u : (u | 0x80000000u);
}
__device__ __forceinline__ float unflip_f(unsigned u) {
  unsigned v = (u & 0x80000000u) ? (u & 0x7FFFFFFFu) : ~u;
  return __uint_as_float(v);
}
__device__ __forceinline__ float elu_f(float v) {
  return v > 0.f ? v : (expf(v) - 1.f);
}

// -------------------- utility --------------------
__global__ void k_zero(float* p, long long n) {
  long long i = (long long)blockIdx.x * blockDim.x + threadIdx.x;
  if (i < n) p[i] = 0.f;
}

// sensor flags + masked global sum (gsum[0..9] = sum x over sensors, gsum[10] = count)
__global__ void k_pool(const float* __restrict__ x, const float* __restrict__ known,
                       float* __restrict__ sf, float* __restrict__ gsum) {
  int n = blockIdx.x * blockDim.x + threadIdx.x;
  if (n >= N_NODES) return;
  float s = known[n * DIN + (DIN - 1)] > 0.f ? 1.f : 0.f;
  sf[n] = s;
  if (s > 0.f) {
    for (int d = 0; d < DIN; ++d) atomicAdd(&gsum[d], x[n * DIN + d]);
    atomicAdd(&gsum[DIN], 1.f);
  }
}

// tiny 2-layer MLP on the pooled vector: g = elu(elu(g@gw1+gb1)@gw2+gb2)  (1 block, 64 threads)
__global__ void k_gmlp(const float* __restrict__ gsum,
                       const float* __restrict__ gw1, const float* __restrict__ gb1,
                       const float* __restrict__ gw2, const float* __restrict__ gb2,
                       float* __restrict__ g) {
  __shared__ float gin[DIN];
  __shared__ float g1[C64];
  int t = threadIdx.x;
  float cnt = fmaxf(gsum[DIN], 1.f);
  if (t < DIN) gin[t] = gsum[t] / cnt;
  __syncthreads();
  float a = gb1[t];
  for (int d = 0; d < DIN; ++d) a += gin[d] * gw1[d * C64 + t];
  g1[t] = elu_f(a);
  __syncthreads();
  float b = gb2[t];
  for (int j = 0; j < C64; ++j) b += g1[j] * gw2[j * C64 + t];
  g[t] = elu_f(b);
}

// h = concat(x, known, broadcast(g))  -> [N, 84]
__global__ void k_buildh(const float* __restrict__ x, const float* __restrict__ known,
                         const float* __restrict__ g, float* __restrict__ h) {
  long long i = (long long)blockIdx.x * blockDim.x + threadIdx.x;
  if (i >= (long long)N_NODES * D1) return;
  int n = (int)(i / D1), k = (int)(i % D1);
  float v;
  if (k < DIN)            v = x[n * DIN + k];
  else if (k < 2 * DIN)   v = known[n * DIN + (k - DIN)];
  else                    v = g[k - 2 * DIN];
  h[i] = v;
}

// -------- fp32 WMMA GEMM: Y[N x 64] = H[N x Din] @ W[Din x 64] (+bias) --------
// 128 threads = 4 waves; wave w owns column tile w (cols 16w..16w+15), block owns 16 rows.
// A(16x4) layout: lanes 0-15 M=lane K={k0,k0+1}; lanes 16-31 M=lane-16 K={k0+2,k0+3}.
// B(4x16) layout mirrors (VGPR0: K=k0 / k0+2, VGPR1: K=k0+1 / k0+3, N=lane&15).
// C/D(16x16): VGPR v -> M = v + 8*(lane>=16), N = lane&15.
__global__ void k_mm_wmma(const float* __restrict__ H, const float* __restrict__ W,
                          const float* __restrict__ bias, float* __restrict__ Y, int Din) {
  int wave = threadIdx.x >> 5;
  int lane = threadIdx.x & 31;
  int l16  = lane & 15;
  int half = lane >> 4;
  int row0 = blockIdx.x * 16;
  int col  = wave * 16 + l16;
  const float* Arow = H + (long long)(row0 + l16) * Din;
  v8f c = {};
  for (int k0 = 0; k0 < Din; k0 += 4) {
    int kk = k0 + 2 * half;
    v2f a, b;
    a.x = Arow[kk];
    a.y = Arow[kk + 1];
    b.x = W[(long long)kk * C64 + col];
    b.y = W[(long long)(kk + 1) * C64 + col];
    c = __builtin_amdgcn_wmma_f32_16x16x4_f32(false, a, false, b, (short)0, c, false, false);
  }
  float bs = bias ? bias[col] : 0.f;
#pragma unroll
  for (int v = 0; v < 8; ++v) {
    int row = row0 + v + 8 * half;
    Y[(long long)row * C64 + col] = c[v] + bs;
  }
}

// -------- sweep 1: per-edge GATv2 logit + atomic segment max (wave per edge) --------
__global__ void k_edge_logit(const int* __restrict__ src, const int* __restrict__ dst,
                             const float* __restrict__ eattr,
                             const float* __restrict__ xl, const float* __restrict__ xr,
                             const float* __restrict__ we, const float* __restrict__ att,
                             float* __restrict__ elog, unsigned* __restrict__ mx) {
  int e = blockIdx.x * 8 + (threadIdx.x >> 5);
  if (e >= N_EDGES) return;
  int lane = threadIdx.x & 31;
  int s = src[e], d = dst[e];
  float ea = eattr[e];
  float2 a = ((const float2*)(xl + (long long)s * C64))[lane];
  float2 b = ((const float2*)(xr + (long long)d * C64))[lane];
  float2 w = ((const float2*)we)[lane];
  float2 t = ((const float2*)att)[lane];
  float m0 = a.x + b.x + ea * w.x;
  float m1 = a.y + b.y + ea * w.y;
  m0 = m0 > 0.f ? m0 : 0.2f * m0;      // leaky_relu(0.2)
  m1 = m1 > 0.f ? m1 : 0.2f * m1;
  float p = m0 * t.x + m1 * t.y;
  for (int off = 16; off > 0; off >>= 1) p += __shfl_xor(p, off, 32);
  if (lane == 0) {
    elog[e] = p;
    atomicMax(&mx[d], flip_f(p));      // max over ALL edges (mask applied later, as in ref)
  }
}

// -------- sweep 2: exp(logit - max) * mask, atomic denominator (thread per edge) --------
__global__ void k_edge_soft(const int* __restrict__ src, const int* __restrict__ dst,
                            const float* __restrict__ sf, const unsigned* __restrict__ mx,
                            float* __restrict__ elog, float* __restrict__ den, int mode) {
  int e = blockIdx.x * blockDim.x + threadIdx.x;
  if (e >= N_EDGES) return;
  int s = src[e], d = dst[e];
  float m;
  if (mode == 1)      m = sf[s] * sf[d];
  else if (mode == 2) { float a = sf[s], b = sf[d]; m = a + b - 2.f * a * b; }
  else                m = 1.f;
  float ex = 0.f;
  if (m != 0.f) ex = expf(elog[e] - unflip_f(mx[d])) * m;
  elog[e] = ex;
  if (ex != 0.f) atomicAdd(&den[d], ex);
}

// -------- sweep 3: acc[dst] += alpha * xl[src]  (wave per edge, 2 channels/lane) --------
__global__ void k_edge_agg(const int* __restrict__ src, const int* __restrict__ dst,
                           const float* __restrict__ elog, const float* __restrict__ den,
                           const float* __restrict__ xl, float* __restrict__ acc) {
  int e = blockIdx.x * 8 + (threadIdx.x >> 5);
  if (e >= N_EDGES) return;
  float ex = elog[e];
  if (ex == 0.f) return;
  int lane = threadIdx.x & 31;
  int d = dst[e], s = src[e];
  float alpha = ex / (den[d] + 1e-16f);
  float2 xv = ((const float2*)(xl + (long long)s * C64))[lane];
  atomicAdd(&acc[(long long)d * C64 + lane * 2 + 0], alpha * xv.x);
  atomicAdd(&acc[(long long)d * C64 + lane * 2 + 1], alpha * xv.y);
}

// -------- epilogue: out = elu(acc + bias) + resid  (elementwise; in-place safe) --------
__global__ void k_epi_res(const float* __restrict__ acc, const float* __restrict__ bias,
                          const float* __restrict__ resid, float* __restrict__ out) {
  long long i = (long long)blockIdx.x * blockDim.x + threadIdx.x;
  if (i >= (long long)N_NODES * C64) return;
  int cidx = (int)(i & 63);
  out[i] = elu_f(acc[i] + bias[cidx]) + resid[i];
}

// -------- layer-3 epilogue fused with readout: y[n] = (acc[n]+bias) . ro_w + ro_b --------
__global__ void k_readout(const float* __restrict__ acc, const float* __restrict__ bias,
                          const float* __restrict__ ro_w, const float* __restrict__ ro_b,
                          float* __restrict__ out) {
  int n = blockIdx.x * 8 + (threadIdx.x >> 5);
  if (n >= N_NODES) return;
  int lane = threadIdx.x & 31;
  float2 av = ((const float2*)(acc + (long long)n * C64))[lane];
  float2 bv = ((const float2*)bias)[lane];
  float2 rv = ((const float2*)ro_w)[lane];
  float p = (av.x + bv.x) * rv.x + (av.y + bv.y) * rv.y;
  for (int off = 16; off > 0; off >>= 1) p += __shfl_xor(p, off, 32);
  if (lane == 0) out[n] = p + ro_b[0];
}

extern "C" void kernel_launch(void* const* d_in, const int* in_sizes, int n_in,
                              void* d_out, int out_size, void* d_ws, size_t ws_size,
                              hipStream_t stream) {
  const float* x     = (const float*)d_in[0];
  const float* known = (const float*)d_in[1];
  const int*   eidx  = (const int*)d_in[2];
  const float* eattr = (const float*)d_in[3];
  const float* gw1   = (const float*)d_in[4];
  const float* gb1   = (const float*)d_in[5];
  const float* gw2   = (const float*)d_in[6];
  const float* gb2   = (const float*)d_in[7];
  // conv1 @ 8..13, conv2 @ 14..19, conv3 @ 20..25: {wl, bl, wr, we, att, bias}
  const float* res_w = (const float*)d_in[26];
  const float* res_b = (const float*)d_in[27];
  const float* ro_w  = (const float*)d_in[28];
  const float* ro_b  = (const float*)d_in[29];
  const int* src = eidx;
  const int* dst = eidx + N_EDGES;

  // workspace carve-up (floats)
  float* ws   = (float*)d_ws;
  float* gsum = ws;                                   // 16
  float* gvec = ws + 16;                              // 64
  float* sf   = ws + 80;                              // N
  float* h    = sf + N_NODES;                         // N*84
  float* xl   = h  + (long long)N_NODES * D1;         // N*64
  float* xr   = xl + (long long)N_NODES * C64;        // N*64
  float* res  = xr + (long long)N_NODES * C64;        // N*64
  float* acc  = res + (long long)N_NODES * C64;       // N*64
  float* cur  = acc + (long long)N_NODES * C64;       // N*64
  unsigned* mxu = (unsigned*)(cur + (long long)N_NODES * C64);  // N (contig with den)
  float* den  = (float*)(mxu + N_NODES);              // N
  float* elog = den + N_NODES;                        // E

  // ---- global pooling + context vector + h build ----
  k_zero<<<1, 16, 0, stream>>>(gsum, 16);
  k_pool<<<(N_NODES + 255) / 256, 256, 0, stream>>>(x, known, sf, gsum);
  k_gmlp<<<1, 64, 0, stream>>>(gsum, gw1, gb1, gw2, gb2, gvec);
  long long nh = (long long)N_NODES * D1;
  k_buildh<<<(unsigned)((nh + 255) / 256), 256, 0, stream>>>(x, known, gvec, h);

  // residual projection res = h @ res_w + res_b   (WMMA)
  k_mm_wmma<<<N_NODES / 16, 128, 0, stream>>>(h, res_w, res_b, res, D1);

  const float* hin = h;
  int din = D1;
  long long nacc = (long long)N_NODES * C64;
  for (int layer = 0; layer < 3; ++layer) {
    int pb = 8 + layer * 6;
    const float* wl   = (const float*)d_in[pb + 0];
    const float* bl   = (const float*)d_in[pb + 1];
    const float* wr   = (const float*)d_in[pb + 2];
    const float* we   = (const float*)d_in[pb + 3];
    const float* att  = (const float*)d_in[pb + 4];
    const float* bias = (const float*)d_in[pb + 5];

    // node transforms (WMMA fp32)
    k_mm_wmma<<<N_NODES / 16, 128, 0, stream>>>(hin, wl, bl, xl, din);
    k_mm_wmma<<<N_NODES / 16, 128, 0, stream>>>(hin, wr, nullptr, xr, din);

    // clear accumulators (acc) and segment max/denominator (mxu|den contiguous)
    k_zero<<<(unsigned)((nacc + 255) / 256), 256, 0, stream>>>(acc, nacc);
    k_zero<<<(2 * N_NODES + 255) / 256, 256, 0, stream>>>((float*)mxu, 2 * N_NODES);

    // segment softmax + aggregation
    k_edge_logit<<<(N_EDGES + 7) / 8, 256, 0, stream>>>(src, dst, eattr, xl, xr, we, att, elog, mxu);
    int mode = (layer == 0) ? 1 : (layer == 1) ? 2 : 3;
    k_edge_soft<<<(N_EDGES + 255) / 256, 256, 0, stream>>>(src, dst, sf, mxu, elog, den, mode);
    k_edge_agg<<<(N_EDGES + 7) / 8, 256, 0, stream>>>(src, dst, elog, den, xl, acc);

    // epilogue
    if (layer == 0) {
      k_epi_res<<<(unsigned)((nacc + 255) / 256), 256, 0, stream>>>(acc, bias, res, cur);
      hin = cur; din = C64;
    } else if (layer == 1) {
      k_epi_res<<<(unsigned)((nacc + 255) / 256), 256, 0, stream>>>(acc, bias, cur, cur);
    } else {
      k_readout<<<(N_NODES + 7) / 8, 256, 0, stream>>>(acc, bias, ro_w, ro_b, (float*)d_out);
    }
  }
}